// PathEncoder_12584254177665
// MI455X (gfx1250) — compile-verified
//
#include <hip/hip_runtime.h>
#include <math.h>

typedef __attribute__((ext_vector_type(2))) float v2f;
typedef __attribute__((ext_vector_type(8))) float v8f;

// Problem shape (fixed by the harness/reference)
constexpr int B = 16, N = 128, E = 2048, L = 4, H = 8, D = 64;
constexpr int KTOT  = L * D;        // 256 = contraction length
constexpr int TILE  = 16;           // positions (M rows) per wave
constexpr int WAVES = 2;            // waves per block
constexpr int ASTRIDE = KTOT + 4;   // 260: lane banks 4m / 4m+2 -> conflict-free,
                                    // rows stay 16B-aligned for b128 stores

__global__ __launch_bounds__(WAVES * 32)
void path_encoder_wmma(const float* __restrict__ edge_feat,   // (B,E,D)
                       const int*   __restrict__ path_idx,    // (B,N,N,L)
                       const int*   __restrict__ dist,        // (B,N,N)
                       const int*   __restrict__ node_counts, // (B,)
                       const float* __restrict__ emb_table,   // (L*H,D)
                       float* __restrict__ out)               // (B,N,N,H)
{
    __shared__ float ldsA[WAVES][TILE * ASTRIDE];   // 2 * 16*260*4 = 33280 B
    __shared__ float ldsB[(KTOT / 2) * 32];         // 128*32*4     = 16384 B
    __shared__ int   ldsIdx[WAVES][TILE * L];       //               512 B

    const int tid  = threadIdx.x;
    const int wv   = tid >> 5;
    const int lane = tid & 31;
    const int m    = lane & 15;     // row/col index within 16
    const int hi   = lane >> 4;     // half-wave select

    const int tileId  = blockIdx.x * WAVES + wv;        // [0, B*N*N/16)
    const int b       = tileId >> 10;                   // / (N*N/TILE) = /1024
    const int t       = tileId & 1023;
    const int posBase = t * TILE;                       // flat (x,y) base, 16 | N
    const int x       = posBase >> 7;                   // / N
    const int y0      = posBase & (N - 1);

    // ---- Stage B: emb (L,H,D) -> K-major interleaved pairs, zero-padded to 16 cols.
    // ldsB[k2*32 + n*2 + j] = emb[l, n, d] for k = 2*k2 + j = l*64 + d, n < H else 0.
    for (int e2 = tid; e2 < (KTOT / 2) * 16; e2 += WAVES * 32) {
        const int k2 = e2 >> 4, n = e2 & 15;
        float2 v = make_float2(0.f, 0.f);
        if (n < H) {
            const int l = k2 >> 5;          // k = 2*k2 -> l = k/64
            const int d = (k2 << 1) & 63;   // d even -> 8B-aligned pair
            v = *(const float2*)&emb_table[(l * H + n) * D + d];
        }
        *(float2*)&ldsB[k2 * 32 + n * 2] = v;
    }

    // ---- Stage the 64 path indices of this tile (contiguous: pos-major, l inner).
    {
        const int* p = path_idx + ((size_t)b * N * N + posBase) * L;
        ldsIdx[wv][lane]      = __builtin_nontemporal_load(p + lane);
        ldsIdx[wv][lane + 32] = __builtin_nontemporal_load(p + lane + 32);
    }
    __syncthreads();

    // ---- Gather A: 64 rows of 64 f32 from edge_feat; idx==E -> zeros (padding slot).
    // Two rows per iteration: 16 lanes * float4 = one coalesced 256B row each.
    const float* ebase = edge_feat + (size_t)b * E * D;
    for (int i = 0; i < 32; ++i) {
        const int r   = 2 * i + hi;        // row in [0,64): r = p*L + l
        const int p   = r >> 2;
        const int l   = r & 3;
        const int idx = ldsIdx[wv][r];
        float4 v = make_float4(0.f, 0.f, 0.f, 0.f);
        if (idx < E) {
            v = *(const float4*)(ebase + idx * D + (m << 2));  // 32-bit offset form
        }
        *(float4*)&ldsA[wv][p * ASTRIDE + l * D + (m << 2)] = v;
    }
    __syncthreads();

    // ---- GEMM: D(16x16) += A(16x256) * B(256x16) via 64x V_WMMA_F32_16X16X4_F32.
    // A frag: lane(m,hi) = { A[m, ks+2*hi], A[m, ks+2*hi+1] }
    // B frag: lane(m,hi) = { B[ks+2*hi, m], B[ks+2*hi+1, m] }
    v8f acc = {};
    const float* aBase = &ldsA[wv][m * ASTRIDE + 2 * hi];
#pragma unroll 8
    for (int s = 0; s < KTOT / 4; ++s) {
        v2f a  = *(const v2f*)(aBase + 4 * s);
        v2f bq = *(const v2f*)&ldsB[(2 * s + hi) * 32 + (m << 1)];
        acc = __builtin_amdgcn_wmma_f32_16x16x4_f32(
            /*neg_a=*/false, a, /*neg_b=*/false, bq,
            /*c_mod=*/(short)0, acc, /*reuse_a=*/false, /*reuse_b=*/false);
    }

    // ---- Epilogue. Per-row scale = 1/clamp(dist,1,4). Since the domain is
    // {1,2,3,4}, a cndmask LUT of the correctly-rounded constants is bit-identical
    // to an IEEE f32 division -- no div chain needed. Lanes 0..15 hold the 16 row
    // factors; broadcast via wave32 shuffle. D layout: acc[r] -> C[r + 8*hi, m].
    float fac = 1.0f;
    if (lane < 16) {
        const int dcv = __builtin_nontemporal_load(
            dist + (size_t)b * N * N + posBase + lane);
        fac = (dcv <= 1) ? 1.0f
            : (dcv == 2) ? 0.5f
            : (dcv == 3) ? (1.0f / 3.0f)
                         : 0.25f;          // dcv >= 4 (clamped)
    }
    const int  nc = __builtin_amdgcn_readfirstlane(node_counts[b]);
    const bool xv = (x < nc);
#pragma unroll
    for (int r = 0; r < 8; ++r) {
        const int   row = r + 8 * hi;
        const float f   = __shfl(fac, row, 32);
        const bool  valid = xv && (y0 + row < nc);
        const float val   = valid ? acc[r] * f : -INFINITY;
        if (m < H) {
            __builtin_nontemporal_store(
                val, out + ((size_t)b * N * N + posBase + row) * H + m);
        }
    }
}

extern "C" void kernel_launch(void* const* d_in, const int* in_sizes, int n_in,
                              void* d_out, int out_size, void* d_ws, size_t ws_size,
                              hipStream_t stream) {
    const float* edge_feat = (const float*)d_in[0];
    const int*   path_idx  = (const int*)d_in[1];
    const int*   dist      = (const int*)d_in[2];
    const int*   node_cnt  = (const int*)d_in[3];
    const float* emb_table = (const float*)d_in[4];
    float*       out       = (float*)d_out;

    const int tiles = B * (N * N / TILE);   // 16384
    dim3 grid(tiles / WAVES);               // 8192 blocks, exact (no tail)
    dim3 block(WAVES * 32);                 // 64 threads = 2 waves
    hipLaunchKernelGGL(path_encoder_wmma, grid, block, 0, stream,
                       edge_feat, path_idx, dist, node_cnt, emb_table, out);
}